// PRMGNN_36266703847568
// MI455X (gfx1250) — compile-verified
//
#include <hip/hip_runtime.h>
#include <math.h>

typedef __attribute__((ext_vector_type(2))) float v2f;
typedef __attribute__((ext_vector_type(8))) float v8f;

#define NEG_SLOPE 0.2f
#define GAT_EPS 1e-16f

__device__ __forceinline__ float leaky(float v) { return v > 0.f ? v : NEG_SLOPE * v; }

// Native hardware f32 atomic add (avoids CAS-loop expansion of plain atomicAdd).
__device__ __forceinline__ void atomAddF32(float* p, float v) {
  unsafeAtomicAdd(p, v);   // lowers to global_atomic_add_f32 on CDNA targets
}

// D[M,Nc] = A[M,K] * B[Nc,K]^T   (A,B,D row-major; B stored [Nc,K] so this is NT GEMM)
// fp32 WMMA 16x16x4, one wave per 16x16 output tile. Requires M%16==0, K%4==0, Nc%16==0.
// A-frag (16x4 f32, ISA 7.12.2): lane<16 -> M=lane, K=0..1 in v0..v1; lane>=16 -> K=2..3.
// B-frag mirrors A with M->N (B rows striped across lanes). C/D: v[r] = row r+8*hi, col lane&15.
__global__ void gat_gemm_nt_f32wmma(const float* __restrict__ A,
                                    const float* __restrict__ B,
                                    float* __restrict__ D,
                                    int K, int Nc, int totalTiles) {
  int wid = (blockIdx.x * blockDim.x + threadIdx.x) >> 5;   // wave id (blockDim multiple of 32)
  if (wid >= totalTiles) return;                            // wave-uniform -> EXEC stays all-1
  int lane = threadIdx.x & 31;
  int colTiles = Nc >> 4;
  int rt = wid / colTiles;
  int ct = wid - rt * colTiles;
  int lr = lane & 15;
  int hi = lane >> 4;
  const float* arow = A + (size_t)(rt * 16 + lr) * K + 2 * hi;
  const float* brow = B + (size_t)(ct * 16 + lr) * K + 2 * hi;
  v8f acc = {0.f, 0.f, 0.f, 0.f, 0.f, 0.f, 0.f, 0.f};
#pragma unroll 8
  for (int k = 0; k < K; k += 4) {
    v2f a = *(const v2f*)(arow + k);    // 8B-aligned: (k + 2*hi)*4 bytes
    v2f b = *(const v2f*)(brow + k);
    acc = __builtin_amdgcn_wmma_f32_16x16x4_f32(false, a, false, b, (short)0, acc,
                                                false, false);
  }
  float* dcol = D + (size_t)ct * 16 + lr;
#pragma unroll
  for (int r = 0; r < 8; ++r)
    dcol[(size_t)(rt * 16 + r + 8 * hi) * Nc] = acc[r];
}

// Per-(node,head) attention logit halves for layer 1: h1 index n*256 + h*32 == tid*32.
__global__ void alpha1_kernel(const float* __restrict__ h1,
                              const float* __restrict__ att_s,
                              const float* __restrict__ att_d,
                              float* __restrict__ as1, float* __restrict__ ad1, int total) {
  int tid = blockIdx.x * blockDim.x + threadIdx.x;
  if (tid >= total) return;
  int h = tid & 7;
  const float* row = h1 + (size_t)tid * 32;
  const float* sw = att_s + h * 32;
  const float* dw = att_d + h * 32;
  float a = 0.f, b = 0.f;
#pragma unroll
  for (int c = 0; c < 32; ++c) { float v = row[c]; a += v * sw[c]; b += v * dw[c]; }
  as1[tid] = a;
  ad1[tid] = b;
}

__global__ void edge_denom1_kernel(const int* __restrict__ ei,
                                   const float* __restrict__ as1,
                                   const float* __restrict__ ad1,
                                   float* __restrict__ den1, int E, int Etot) {
  int tid = blockIdx.x * blockDim.x + threadIdx.x;
  if (tid >= Etot * 8) return;
  int e = tid >> 3, h = tid & 7;
  int s, d;
  if (e < E) { s = ei[e]; d = ei[E + e]; } else { s = d = e - E; }   // self-loops appended
  float w = expf(leaky(as1[s * 8 + h] + ad1[d * 8 + h]));
  atomAddF32(&den1[d * 8 + h], w);
}

// One block (256 thr) per edge; thread = one of 256 channels (head = ch>>5).
__global__ void edge_agg1_kernel(const int* __restrict__ ei,
                                 const float* __restrict__ h1,
                                 const float* __restrict__ as1,
                                 const float* __restrict__ ad1,
                                 const float* __restrict__ den1,
                                 float* __restrict__ agg1, int E) {
  int e = blockIdx.x;
  int ch = threadIdx.x;
  int h = ch >> 5;
  int s, d;
  if (e < E) { s = ei[e]; d = ei[E + e]; } else { s = d = e - E; }
  float w = expf(leaky(as1[s * 8 + h] + ad1[d * 8 + h]));
  float alpha = w / (den1[d * 8 + h] + GAT_EPS);
  atomAddF32(&agg1[(size_t)d * 256 + ch], h1[(size_t)s * 256 + ch] * alpha);
}

__global__ void bias_elu1_kernel(float* __restrict__ agg1, const float* __restrict__ b1,
                                 int total) {
  int tid = blockIdx.x * blockDim.x + threadIdx.x;
  if (tid >= total) return;
  float v = agg1[tid] + b1[tid & 255];
  agg1[tid] = v > 0.f ? v : (expf(v) - 1.f);
}

__global__ void alpha2_kernel(const float* __restrict__ h2,
                              const float* __restrict__ att_s,
                              const float* __restrict__ att_d,
                              float* __restrict__ as2, float* __restrict__ ad2, int N) {
  int n = blockIdx.x * blockDim.x + threadIdx.x;
  if (n >= N) return;
  const float* row = h2 + (size_t)n * 64;
  float a = 0.f, b = 0.f;
#pragma unroll
  for (int c = 0; c < 64; ++c) { float v = row[c]; a += v * att_s[c]; b += v * att_d[c]; }
  as2[n] = a;
  ad2[n] = b;
}

__global__ void edge_denom2_kernel(const int* __restrict__ ei,
                                   const float* __restrict__ as2,
                                   const float* __restrict__ ad2,
                                   float* __restrict__ den2, int E, int Etot) {
  int e = blockIdx.x * blockDim.x + threadIdx.x;
  if (e >= Etot) return;
  int s, d;
  if (e < E) { s = ei[e]; d = ei[E + e]; } else { s = d = e - E; }
  atomAddF32(&den2[d], expf(leaky(as2[s] + ad2[d])));
}

__global__ void edge_agg2_kernel(const int* __restrict__ ei,
                                 const float* __restrict__ h2,
                                 const float* __restrict__ as2,
                                 const float* __restrict__ ad2,
                                 const float* __restrict__ den2,
                                 float* __restrict__ agg2, int E, int Etot) {
  int tid = blockIdx.x * blockDim.x + threadIdx.x;
  if (tid >= Etot * 64) return;
  int e = tid >> 6, ch = tid & 63;
  int s, d;
  if (e < E) { s = ei[e]; d = ei[E + e]; } else { s = d = e - E; }
  float w = expf(leaky(as2[s] + ad2[d]));
  float alpha = w / (den2[d] + GAT_EPS);
  atomAddF32(&agg2[(size_t)d * 64 + ch], h2[(size_t)s * 64 + ch] * alpha);
}

// out2 = agg2 + b2 ; logits = out2 @ fc_w^T + fc_b ; log_softmax over the 2 classes.
__global__ void final_kernel(const float* __restrict__ agg2, const float* __restrict__ b2,
                             const float* __restrict__ fcw, const float* __restrict__ fcb,
                             float* __restrict__ out, int N) {
  int n = blockIdx.x * blockDim.x + threadIdx.x;
  if (n >= N) return;
  const float* row = agg2 + (size_t)n * 64;
  float l0 = fcb[0], l1 = fcb[1];
#pragma unroll
  for (int j = 0; j < 64; ++j) {
    float o = row[j] + b2[j];
    l0 += o * fcw[j];
    l1 += o * fcw[64 + j];
  }
  float m = fmaxf(l0, l1);
  float lse = m + logf(expf(l0 - m) + expf(l1 - m));
  out[(size_t)n * 2 + 0] = l0 - lse;
  out[(size_t)n * 2 + 1] = l1 - lse;
}

extern "C" void kernel_launch(void* const* d_in, const int* in_sizes, int n_in,
                              void* d_out, int out_size, void* d_ws, size_t ws_size,
                              hipStream_t stream) {
  const float* x    = (const float*)d_in[0];
  const int*   ei   = (const int*)d_in[1];   // [2,E]: src then dst
  const float* W1   = (const float*)d_in[2]; // [256,128]
  const float* aS1  = (const float*)d_in[3]; // [8,32]
  const float* aD1  = (const float*)d_in[4];
  const float* b1   = (const float*)d_in[5]; // [256]
  const float* W2   = (const float*)d_in[6]; // [64,256]
  const float* aS2  = (const float*)d_in[7]; // [1,64]
  const float* aD2  = (const float*)d_in[8];
  const float* b2   = (const float*)d_in[9]; // [64]
  const float* fcw  = (const float*)d_in[10]; // [2,64]
  const float* fcb  = (const float*)d_in[11]; // [2]
  float* out = (float*)d_out;

  const int IN = 128, F1 = 256, F2 = 64;
  const int N = in_sizes[0] / IN;       // 50000 (divisible by 16)
  const int E = in_sizes[1] / 2;        // 800000
  const int Etot = E + N;               // edges + self-loops

  // Workspace layout (floats). h1 region is reused for layer-2 buffers once free.
  float* w    = (float*)d_ws;
  float* h1   = w;                               // N*F1
  float* agg1 = h1 + (size_t)N * F1;             // N*F1
  float* as1  = agg1 + (size_t)N * F1;           // N*8
  float* ad1  = as1 + (size_t)N * 8;             // N*8
  float* den1 = ad1 + (size_t)N * 8;             // N*8
  float* h2   = h1;                              // N*F2  (reuse: h1 dead after edge_agg1)
  float* agg2 = h1 + (size_t)N * F2;             // N*F2
  float* as2  = h1 + (size_t)2 * N * F2;         // N
  float* ad2  = as2 + N;                         // N
  float* den2 = ad2 + N;                         // N

  // ---- Layer 1 ----
  hipMemsetAsync(den1, 0, (size_t)N * 8 * sizeof(float), stream);
  hipMemsetAsync(agg1, 0, (size_t)N * F1 * sizeof(float), stream);

  {
    int tiles = (N / 16) * (F1 / 16);
    int blocks = (tiles * 32 + 127) / 128;
    gat_gemm_nt_f32wmma<<<blocks, 128, 0, stream>>>(x, W1, h1, IN, F1, tiles);
  }
  alpha1_kernel<<<(N * 8 + 255) / 256, 256, 0, stream>>>(h1, aS1, aD1, as1, ad1, N * 8);
  edge_denom1_kernel<<<(Etot * 8 + 255) / 256, 256, 0, stream>>>(ei, as1, ad1, den1, E, Etot);
  edge_agg1_kernel<<<Etot, 256, 0, stream>>>(ei, h1, as1, ad1, den1, agg1, E);
  bias_elu1_kernel<<<(N * F1 + 255) / 256, 256, 0, stream>>>(agg1, b1, N * F1);

  // ---- Layer 2 ----
  hipMemsetAsync(den2, 0, (size_t)N * sizeof(float), stream);
  hipMemsetAsync(agg2, 0, (size_t)N * F2 * sizeof(float), stream);

  {
    int tiles = (N / 16) * (F2 / 16);
    int blocks = (tiles * 32 + 127) / 128;
    gat_gemm_nt_f32wmma<<<blocks, 128, 0, stream>>>(agg1, W2, h2, F1, F2, tiles);
  }
  alpha2_kernel<<<(N + 255) / 256, 256, 0, stream>>>(h2, aS2, aD2, as2, ad2, N);
  edge_denom2_kernel<<<(Etot + 255) / 256, 256, 0, stream>>>(ei, as2, ad2, den2, E, Etot);
  edge_agg2_kernel<<<(Etot * 64 + 255) / 256, 256, 0, stream>>>(ei, h2, as2, ad2, den2,
                                                                agg2, E, Etot);
  // ---- FC + log_softmax ----
  final_kernel<<<(N + 255) / 256, 256, 0, stream>>>(agg2, b2, fcw, fcb, out, N);
}